// CAM_48507360641093
// MI455X (gfx1250) — compile-verified
//
#include <hip/hip_runtime.h>
#include <hip/hip_bf16.h>

typedef __attribute__((ext_vector_type(2))) float v2f;
typedef __attribute__((ext_vector_type(4))) float v4f;
typedef __attribute__((ext_vector_type(8))) float v8f;
typedef __attribute__((ext_vector_type(4))) int   v4i;

#define GLOBAL_AS __attribute__((address_space(1)))
#define LDS_AS    __attribute__((address_space(3)))

__device__ __forceinline__ void async_g2l_b128(const float* g, float* l) {
  // GLOBAL_LOAD_ASYNC_TO_LDS_B128: per-lane global src -> per-lane LDS dst,
  // tracked by ASYNCcnt. imm offset = 0, cpol = 0.
  __builtin_amdgcn_global_load_async_to_lds_b128(
      (GLOBAL_AS v4i*)g, (LDS_AS v4i*)l, 0, 0);
}

__device__ __forceinline__ void wait_async0() {
  asm volatile("s_wait_asynccnt 0x0" ::: "memory");
}

// C[b] = A[b] (MxK, row-major, lda) * op(B[b])
//   !TRANSB: Bsrc is KxN row-major (ldb = N)
//    TRANSB: Bsrc is NxK row-major (ldb = K), op(B) = Bsrc^T
// Block: 256 threads (8 wave32), tile 128(M) x 64(N), K-step 32,
// double-buffered LDS filled by async global->LDS DMA.
// Wave grid 4(M) x 2(N); each wave computes a 32x32 tile via 2x2 WMMA accums.
// grid = (N/64, M/128, batch)
template <bool TRANSB>
__global__ __launch_bounds__(256) void gemm_f32_wmma(
    const float* __restrict__ A, long long strideA,
    const float* __restrict__ Bsrc, long long strideB,
    float* __restrict__ Cd, long long strideC,
    int K, int lda, int ldb, int ldc) {
  // A tile stored [m][k] (pad 36 -> 16 distinct banks for frag reads).
  __shared__ float As[2][128][36];
  // B tile: !TRANSB -> [k][n] (32x68); TRANSB -> stored untransposed [n][k]
  // (64x36) and indexed transposed at fragment-read time (no LDS scatter).
  constexpr int BR = TRANSB ? 64 : 32;
  constexpr int BC = TRANSB ? 36 : 68;
  __shared__ float Bs[2][BR][BC];

  const int b  = blockIdx.z;
  const int m0 = blockIdx.y * 128;
  const int n0 = blockIdx.x * 64;

  const int tid  = threadIdx.x;
  const int lane = tid & 31;
  const int wave = tid >> 5;           // 0..7
  const int wm   = wave >> 1;          // 0..3 (M)
  const int wn   = wave & 1;           // 0..1 (N)
  const int lo   = lane & 15;
  const int hi   = lane >> 4;          // 0 or 1

  const float* Ab = A    + (long long)b * strideA;
  const float* Bb = Bsrc + (long long)b * strideB;

  // ---- async stage of one 32-wide K-block into LDS buffer `s` ----
  auto stage = [&](int s, int k0) {
    // A tile: 128 x 32 floats = 1024 v4f, 4 per thread
#pragma unroll
    for (int i = 0; i < 4; ++i) {
      int f   = tid + 256 * i;           // 0..1023
      int row = f >> 3;                  // 0..127
      int c4  = (f & 7) << 2;            // 0,4,...,28
      async_g2l_b128(Ab + (long long)(m0 + row) * lda + (k0 + c4),
                     &As[s][row][c4]);
    }
    // B tile: 2048 floats = 512 v4f, 2 per thread
    if (!TRANSB) {
#pragma unroll
      for (int i = 0; i < 2; ++i) {
        int f   = tid + 256 * i;         // 0..511
        int row = f >> 4;                // k-row 0..31
        int c4  = (f & 15) << 2;         // 0..60
        async_g2l_b128(Bb + (long long)(k0 + row) * ldb + (n0 + c4),
                       &Bs[s][row][c4]);
      }
    } else {
#pragma unroll
      for (int i = 0; i < 2; ++i) {
        int f    = tid + 256 * i;        // 0..511
        int nrow = f >> 3;               // 0..63
        int c4   = (f & 7) << 2;         // 0..28
        async_g2l_b128(Bb + (long long)(n0 + nrow) * ldb + (k0 + c4),
                       &Bs[s][nrow][c4]);
      }
    }
  };

  v8f acc[2][2];
  acc[0][0] = (v8f)0.f; acc[0][1] = (v8f)0.f;
  acc[1][0] = (v8f)0.f; acc[1][1] = (v8f)0.f;

  int buf = 0;
  stage(0, 0);

  for (int k0 = 0; k0 < K; k0 += 32) {
    wait_async0();      // this wave's DMA into `buf` has landed in LDS
    __syncthreads();    // ... and every other wave's too
    if (k0 + 32 < K) stage(buf ^ 1, k0 + 32);  // prefetch next block

    // ---- compute: 8 K-steps of 4, 2x2 wmma per step = 32 wmma ----
#pragma unroll
    for (int kk = 0; kk < 32; kk += 4) {
      // A frag 16x4: a.x = A[M=lo][K=2*hi], a.y = A[M=lo][K=2*hi+1]
      v2f afrag[2];
#pragma unroll
      for (int mt = 0; mt < 2; ++mt)
        afrag[mt] = *(const v2f*)&As[buf][wm * 32 + mt * 16 + lo][kk + 2 * hi];
      // B frag 4x16: b.x = B[K=2*hi][N=lo], b.y = B[K=2*hi+1][N=lo]
      v2f bfrag[2];
#pragma unroll
      for (int nt = 0; nt < 2; ++nt) {
        int n = wn * 32 + nt * 16 + lo;
        if (!TRANSB) {
          v2f t;
          t.x = Bs[buf][kk + 2 * hi + 0][n];
          t.y = Bs[buf][kk + 2 * hi + 1][n];
          bfrag[nt] = t;
        } else {
          bfrag[nt] = *(const v2f*)&Bs[buf][n][kk + 2 * hi];
        }
      }
#pragma unroll
      for (int mt = 0; mt < 2; ++mt)
#pragma unroll
        for (int nt = 0; nt < 2; ++nt)
          acc[mt][nt] = __builtin_amdgcn_wmma_f32_16x16x4_f32(
              false, afrag[mt], false, bfrag[nt], (short)0, acc[mt][nt],
              false, false);
    }

    __syncthreads();    // all waves done reading `buf` before it is restaged
    buf ^= 1;
  }

  // ---- epilogue: C/D layout: VGPR v -> M = v + 8*hi, N = lo ----
  float* Cb = Cd + (long long)b * strideC;
#pragma unroll
  for (int mt = 0; mt < 2; ++mt)
#pragma unroll
    for (int nt = 0; nt < 2; ++nt) {
      int mrow = m0 + wm * 32 + mt * 16 + hi * 8;
      int ncol = n0 + wn * 32 + nt * 16 + lo;
#pragma unroll
      for (int v = 0; v < 8; ++v)
        Cb[(long long)(mrow + v) * ldc + ncol] = acc[mt][nt][v];
    }
}

// In-place row softmax over rows of length 512; one 256-thread block per row.
__global__ __launch_bounds__(256) void softmax512(float* __restrict__ E) {
  __shared__ float red[256];
  float* p = E + (long long)blockIdx.x * 512;
  int t = threadIdx.x;
  float a = p[t], b2 = p[t + 256];
  red[t] = fmaxf(a, b2);
  __syncthreads();
#pragma unroll
  for (int s = 128; s > 0; s >>= 1) {
    if (t < s) red[t] = fmaxf(red[t], red[t + s]);
    __syncthreads();
  }
  float rowmax = red[0];
  __syncthreads();
  float ea = __expf(a - rowmax), eb = __expf(b2 - rowmax);
  red[t] = ea + eb;
  __syncthreads();
#pragma unroll
  for (int s = 128; s > 0; s >>= 1) {
    if (t < s) red[t] += red[t + s];
    __syncthreads();
  }
  float inv = 1.0f / red[0];
  p[t] = ea * inv;
  p[t + 256] = eb * inv;
}

extern "C" void kernel_launch(void* const* d_in, const int* in_sizes, int n_in,
                              void* d_out, int out_size, void* d_ws, size_t ws_size,
                              hipStream_t stream) {
  const float* x  = (const float*)d_in[0];   // (16, 512, 64, 64) f32
  const float* Wc = (const float*)d_in[1];   // (512, 512) f32
  float* out = (float*)d_out;                // (16, 512, 64, 64) f32

  const int B = 16, C = 512, HW = 4096;
  const long long xbs = (long long)C * HW;   // per-batch X stride (elements)
  const long long cbs = (long long)C * C;    // per-batch 512x512 stride

  // Workspace: G = X X^T, T = Wc G, E = T Wc^T (softmax in place) : 48 MiB
  float* G = (float*)d_ws;
  float* T = G + (long long)B * cbs;
  float* E = T + (long long)B * cbs;

  dim3 blk(256);

  // 1) G[b] = X[b] * X[b]^T          (M=512, N=512, K=4096)
  gemm_f32_wmma<true><<<dim3(C / 64, C / 128, B), blk, 0, stream>>>(
      x, xbs, x, xbs, G, cbs, HW, HW, HW, C);

  // 2) T[b] = Wc * G[b]              (M=512, N=512, K=512), Wc broadcast
  gemm_f32_wmma<false><<<dim3(C / 64, C / 128, B), blk, 0, stream>>>(
      Wc, 0, G, cbs, T, cbs, C, C, C, C);

  // 3) E[b] = T[b] * Wc^T            (M=512, N=512, K=512), Wc broadcast
  gemm_f32_wmma<true><<<dim3(C / 64, C / 128, B), blk, 0, stream>>>(
      T, cbs, Wc, 0, E, cbs, C, C, C, C);

  // 4) attention = softmax(E) rows, in place
  softmax512<<<dim3(B * C), blk, 0, stream>>>(E);

  // 5) out[b] = E[b] * X[b]          (M=512, N=4096, K=512)
  gemm_f32_wmma<false><<<dim3(HW / 64, C / 128, B), blk, 0, stream>>>(
      E, cbs, x, xbs, out, xbs, C, C, HW, HW);
}